// AttentionModel_35201551958676
// MI455X (gfx1250) — compile-verified
//
#include <hip/hip_runtime.h>

// ---------------- problem constants (from reference) ----------------
#define N_NODES_C 50000
#define D_FEAT_C 128
#define NEG_SLOPE_C 0.2f

typedef float v2f __attribute__((ext_vector_type(2)));
typedef float v8f __attribute__((ext_vector_type(8)));

// Float atomic-max via signed-max (val>=0) / unsigned-min (val<0) bit trick.
// Location initialized to -inf bits (0xFF800000): huge-negative as signed,
// huge as unsigned, so both paths make monotonic progress toward the max.
__device__ __forceinline__ void atomic_max_float(float* addr, float val) {
  if (val >= 0.0f) {
    atomicMax(reinterpret_cast<int*>(addr), __float_as_int(val));
  } else {
    atomicMin(reinterpret_cast<unsigned int*>(addr), __float_as_uint(val));
  }
}

__global__ void init_segments(float* __restrict__ segmax,
                              float* __restrict__ segsum, int n) {
  int i = blockIdx.x * blockDim.x + threadIdx.x;
  if (i < n) {
    reinterpret_cast<int*>(segmax)[i] = (int)0xFF800000u; // -inf
    segsum[i] = 0.0f;
  }
}

// ------------- kernel 1: latent = LeakyReLU(x @ W + b), seg_max -------------
// Tile: 64 edges x 128 features per 128-thread (4-wave) block.
// LDS row stride 132 dwords: float4-aligned and bank-conflict-free for the
// per-lane 8B A-fragment reads (banks = 4*row + k .. all 64 distinct).
#define TILE_ROWS 64
#define LDS_STRIDE 132

__global__ __launch_bounds__(128) void gemv_lrelu_max(
    const float* __restrict__ x, const float* __restrict__ W,
    const float* __restrict__ b, const int* __restrict__ index,
    float* __restrict__ latent, float* __restrict__ segmax) {
  __shared__ float tile[TILE_ROWS * LDS_STRIDE];
  __shared__ float wsh[D_FEAT_C];

  const int tid = threadIdx.x;
  const long long edgeBase = (long long)blockIdx.x * TILE_ROWS;

  if (tid < D_FEAT_C) wsh[tid] = W[tid];

  // Cooperative, fully coalesced tile load (float4 granularity):
  // 64 rows x 32 float4 = 2048 slots, 16 per thread.
  {
    const float4* __restrict__ xv = reinterpret_cast<const float4*>(x);
#pragma unroll
    for (int i = 0; i < 16; ++i) {
      int f = i * 128 + tid;  // float4 slot in [0, 2048)
      int row = f >> 5;       // 32 float4 per row
      int c4 = f & 31;
      float4 v = xv[(edgeBase + row) * (D_FEAT_C / 4) + c4];
      *reinterpret_cast<float4*>(&tile[row * LDS_STRIDE + c4 * 4]) = v;
    }
  }
  __syncthreads();

  const int wave = tid >> 5;
  const int lane = tid & 31;
  const int half = lane >> 4;  // 0: K slots {0,1}; 1: K slots {2,3}
  const int mrow = lane & 15;  // A-matrix row M within the wave's 16-row slab
  const int r0 = wave * 16;

  // A 16x4 f32 layout (ISA 7.12.2): lane L<16 -> M=L, K={kb,kb+1};
  // lane L>=16 -> M=L-16, K={kb+2,kb+3}. B mirrors with N=lane%16; we
  // broadcast W[k] across all N so every D column equals the dot product.
  v8f acc = {};
#pragma unroll
  for (int kb = 0; kb < D_FEAT_C; kb += 4) {
    const int k = kb + 2 * half;
    v2f a = *reinterpret_cast<const v2f*>(&tile[(r0 + mrow) * LDS_STRIDE + k]);
    v2f bw = *reinterpret_cast<const v2f*>(&wsh[k]);
    acc = __builtin_amdgcn_wmma_f32_16x16x4_f32(
        /*neg_a=*/false, a, /*neg_b=*/false, bw,
        /*c_mod=*/(short)0, acc, /*reuse_a=*/false, /*reuse_b=*/false);
  }

  // D layout: VGPR v, lanes 0-15 -> M=v, lanes 16-31 -> M=v+8; columns all
  // identical, so lanes 0 and 16 each emit 8 edges.
  if (mrow == 0) {
    const float bias = b[0];
    const long long ebase = edgeBase + r0 + (half ? 8 : 0);
#pragma unroll
    for (int v = 0; v < 8; ++v) {
      float lat = acc[v] + bias;
      lat = (lat >= 0.0f) ? lat : NEG_SLOPE_C * lat;
      latent[ebase + v] = lat;
      atomic_max_float(&segmax[index[ebase + v]], lat);
    }
  }
}

// ------------- kernel 2: e = exp(latent - seg_max), seg_sum += e -------------
__global__ void exp_scatter_sum(float* __restrict__ lat,
                                const int* __restrict__ index,
                                const float* __restrict__ segmax,
                                float* __restrict__ segsum, int n) {
  int e = blockIdx.x * blockDim.x + threadIdx.x;
  if (e < n) {
    int seg = index[e];
    float v = expf(lat[e] - segmax[seg]);
    lat[e] = v;
    atomicAdd(&segsum[seg], v);
  }
}

// ------------- kernel 3: out = e / seg_sum[idx] -------------
__global__ void normalize(float* __restrict__ out,
                          const int* __restrict__ index,
                          const float* __restrict__ segsum, int n) {
  int e = blockIdx.x * blockDim.x + threadIdx.x;
  if (e < n) {
    out[e] = out[e] / segsum[index[e]];
  }
}

extern "C" void kernel_launch(void* const* d_in, const int* in_sizes, int n_in,
                              void* d_out, int out_size, void* d_ws, size_t ws_size,
                              hipStream_t stream) {
  const float* x = (const float*)d_in[0];     // [E, 128]
  const float* W = (const float*)d_in[1];     // [128, 1]
  const float* b = (const float*)d_in[2];     // [1]
  const int* index = (const int*)d_in[3];     // [E]
  float* out = (float*)d_out;                 // [E, 1] (also holds latent/e)

  const int E = in_sizes[3];
  const int N = N_NODES_C;

  float* segmax = (float*)d_ws;
  float* segsum = segmax + N;

  init_segments<<<(N + 255) / 256, 256, 0, stream>>>(segmax, segsum, N);
  gemv_lrelu_max<<<E / TILE_ROWS, 128, 0, stream>>>(x, W, b, index, out, segmax);
  exp_scatter_sum<<<(E + 255) / 256, 256, 0, stream>>>(out, index, segmax, segsum, E);
  normalize<<<(E + 255) / 256, 256, 0, stream>>>(out, index, segsum, E);
}